// RMultiHeadGraphAttention2_52716428591536
// MI455X (gfx1250) — compile-verified
//
#include <hip/hip_runtime.h>

// ---------------------------------------------------------------------------
// RMultiHeadGraphAttention2 for MI455X (gfx1250, wave32)
//
// logit_i(src) = h[src] . (c_i * a_src[i]),  c_i = cumprod_{j<i}(w[j])
// => node scores for all 4 heads = one f32 GEMM [N x 128] x [128 x 16]
//    via V_WMMA_F32_16X16X4_F32 (exact f32, matches reference).
// Edge phase: 4-float score lookup + exp + f32 atomic scatter; h (25.6 MB),
// out (102 MB) and all score tables are L2-resident (192 MB L2), so the
// ceiling is L2 atomic throughput, not HBM.
// ---------------------------------------------------------------------------

typedef __attribute__((ext_vector_type(2))) float v2f;
typedef __attribute__((ext_vector_type(8))) float v8f;

#define F_OUT 128
#define N_HEAD 4

// ---- Kernel 1: cumulative scales + WMMA-packed attention matrices ---------
// cs:  [4][128]   cumulative per-head diagonal scale c_i
// BsrcP/BdstP: [32 steps][32 lanes][2]  B fragments in WMMA register order:
//   lane l, step s holds B[k][l&15], B[k+1][l&15] with k = s*4 + (l>>4)*2.
__global__ void rgat_prep(const float* __restrict__ w,
                          const float* __restrict__ a,
                          float* __restrict__ cs,
                          float* __restrict__ BsrcP,
                          float* __restrict__ BdstP) {
  int f = threadIdx.x;  // 0..127
  float c1 = w[f];
  float c2 = c1 * w[F_OUT + f];
  float c3 = c2 * w[2 * F_OUT + f];
  cs[f] = 1.0f;  cs[F_OUT + f] = c1;  cs[2 * F_OUT + f] = c2;  cs[3 * F_OUT + f] = c3;

  for (int idx = f; idx < 32 * 64; idx += F_OUT) {
    int s    = idx >> 6;        // K-step
    int l    = (idx >> 1) & 31; // lane
    int half = idx & 1;
    int col  = l & 15;          // head (cols 4..15 zero)
    int k    = s * 4 + ((l >> 4) << 1) + half;
    float bs = 0.0f, bd = 0.0f;
    if (col < 4) {
      float c = 1.0f;
      for (int j = 0; j < col; ++j) c *= w[j * F_OUT + k];  // c_col[k]
      bs = c * a[col * 256 + k];           // c_col * a_src[col][k]
      bd = a[col * 256 + F_OUT + k];       // a_dst[col][k]
    }
    BsrcP[idx] = bs;
    BdstP[idx] = bd;
  }
}

// ---- Kernel 2: S[Mpad x 16] = X[M x 128] * B[128 x 16] via WMMA f32 -------
// One wave per 16-row tile. Row index clamped (not predicated): out-of-range
// rows compute garbage that lands in workspace padding, never read.
__global__ void rgat_score(const float* __restrict__ X,
                           const float* __restrict__ Bp,
                           float* __restrict__ S, int M) {
  int tile = blockIdx.x * (blockDim.x >> 5) + (threadIdx.x >> 5);
  int base = tile * 16;
  if (base >= M) return;               // wave-uniform -> EXEC all-1s below

  int lane = threadIdx.x & 31;
  int col  = lane & 15;
  int kb   = (lane >> 4) * 2;
  int row  = base + (lane & 15);
  row = row < M ? row : M - 1;         // clamp, no exec divergence
  const float* xr = X + (size_t)row * F_OUT;
  const float* bp = Bp + lane * 2;

  v8f acc = {};
#pragma unroll
  for (int s = 0; s < 32; ++s) {
    v2f av = *(const v2f*)(xr + s * 4 + kb);   // global_load_b64
    v2f bv = *(const v2f*)(bp + s * 64);       // coalesced global_load_b64
    acc = __builtin_amdgcn_wmma_f32_16x16x4_f32(
        /*neg_a=*/false, av, /*neg_b=*/false, bv,
        /*c_mod=*/(short)0, acc, /*reuse_a=*/false, /*reuse_b=*/false);
  }

  // C layout: VGPR v -> M = v + (lane>>4)*8, N = lane&15. S rows are padded
  // to a multiple of 16, so store unconditionally.
  int mhalf = (lane >> 4) * 8;
#pragma unroll
  for (int v = 0; v < 8; ++v)
    S[(size_t)(base + mhalf + v) * 16 + col] = acc[v];
}

// ---- Kernel 3: per-edge attention + atomic scatter ------------------------
// One wave per edge, grid-stride so per-lane scale vectors load once.
// Edge scalars forced into SGPRs -> scalar loads + saddr-form atomics.
__global__ void rgat_edge(const int* __restrict__ A,
                          const float* __restrict__ h,
                          const float* __restrict__ inputr,
                          const float* __restrict__ cs,
                          const float* __restrict__ Snode,
                          const float* __restrict__ Srel,
                          float* __restrict__ rowsum,
                          float* __restrict__ out,
                          int E, int N) {
  const int lane = threadIdx.x & 31;
  const int wid  = (blockIdx.x * blockDim.x + threadIdx.x) >> 5;
  const int nw   = (gridDim.x * blockDim.x) >> 5;

  // per-lane diagonal scales (loaded once, L0-resident)
  const float4 cv0 = *(const float4*)(cs + 0 * F_OUT + lane * 4);
  const float4 cv1 = *(const float4*)(cs + 1 * F_OUT + lane * 4);
  const float4 cv2 = *(const float4*)(cs + 2 * F_OUT + lane * 4);
  const float4 cv3 = *(const float4*)(cs + 3 * F_OUT + lane * 4);
  const float4 cv[N_HEAD] = {cv0, cv1, cv2, cv3};

  for (int ev = wid; ev < E; ev += nw) {
    const int e   = __builtin_amdgcn_readfirstlane(ev);
    const int dst = A[e];
    const int rl  = A[E + e];
    const int src = A[2 * E + e];

    const float4 hs = *(const float4*)(h + (size_t)src * F_OUT + lane * 4);
    const float4 rf = *(const float4*)(inputr + (size_t)rl * F_OUT + lane * 4);
    const float4 sn = *(const float4*)(Snode + (size_t)src * 16);
    const float4 sr = *(const float4*)(Srel + (size_t)rl * 16);

    const float logit[N_HEAD] = {sn.x + sr.x, sn.y + sr.y, sn.z + sr.z, sn.w + sr.w};
    float ee[N_HEAD];
#pragma unroll
    for (int i = 0; i < N_HEAD; ++i) {
      float lk = logit[i] > 0.0f ? logit[i] : 0.2f * logit[i];  // leaky_relu
      ee[i] = __expf(-lk);
    }

    if (lane == 0) {  // single exec toggle for all 4 rowsum adds
#pragma unroll
      for (int i = 0; i < N_HEAD; ++i)
        __hip_atomic_fetch_add(&rowsum[(size_t)i * N + dst], ee[i],
                               __ATOMIC_RELAXED, __HIP_MEMORY_SCOPE_AGENT);
    }

#pragma unroll
    for (int i = 0; i < N_HEAD; ++i) {
      float* op = out + ((size_t)i * N + dst) * F_OUT + lane * 4;
      __hip_atomic_fetch_add(&op[0], (cv[i].x * hs.x - rf.x) * ee[i],
                             __ATOMIC_RELAXED, __HIP_MEMORY_SCOPE_AGENT);
      __hip_atomic_fetch_add(&op[1], (cv[i].y * hs.y - rf.y) * ee[i],
                             __ATOMIC_RELAXED, __HIP_MEMORY_SCOPE_AGENT);
      __hip_atomic_fetch_add(&op[2], (cv[i].z * hs.z - rf.z) * ee[i],
                             __ATOMIC_RELAXED, __HIP_MEMORY_SCOPE_AGENT);
      __hip_atomic_fetch_add(&op[3], (cv[i].w * hs.w - rf.w) * ee[i],
                             __ATOMIC_RELAXED, __HIP_MEMORY_SCOPE_AGENT);
    }
  }
}

// ---- Kernel 4: normalize by rowsum ----------------------------------------
__global__ void rgat_norm(float* __restrict__ out,
                          const float* __restrict__ rowsum,
                          int total) {
  int idx = blockIdx.x * blockDim.x + threadIdx.x;
  if (idx >= total) return;
  out[idx] = out[idx] / rowsum[idx >> 7];   // idx>>7 = head*N + node
}

// ---------------------------------------------------------------------------
extern "C" void kernel_launch(void* const* d_in, const int* in_sizes, int n_in,
                              void* d_out, int out_size, void* d_ws, size_t ws_size,
                              hipStream_t stream) {
  const float* h      = (const float*)d_in[0];   // [N,128]
  const float* inputr = (const float*)d_in[1];   // [R,128]
  const float* w      = (const float*)d_in[2];   // [3,1,128]
  const float* a      = (const float*)d_in[3];   // [4,2,128,1]
  const int*   A      = (const int*)d_in[4];     // [3,E]

  const int N    = in_sizes[0] / F_OUT;
  const int NREL = in_sizes[1] / F_OUT;
  const int E    = in_sizes[4] / 3;

  // Workspace (floats): cs | BsrcP | BdstP | Srel(pad) | Snode(pad) | rowsum
  float* ws = (float*)d_ws;
  float* cs    = ws;                      // 512
  float* BsrcP = ws + 512;                // 2048
  float* BdstP = ws + 512 + 2048;         // 2048
  size_t off   = 512 + 2048 + 2048;
  const int nrel_tiles = (NREL + 15) / 16;
  const int node_tiles = (N + 15) / 16;
  float* Srel  = ws + off;  off += (size_t)nrel_tiles * 16 * 16;
  float* Snode = ws + off;  off += (size_t)node_tiles * 16 * 16;
  float* rowsum = ws + off; off += (size_t)N_HEAD * N;
  (void)ws_size;  // total ~4.1 MB

  // Atomic accumulation targets must start at zero every call.
  hipMemsetAsync(d_out, 0, (size_t)out_size * sizeof(float), stream);
  hipMemsetAsync(rowsum, 0, (size_t)N_HEAD * N * sizeof(float), stream);

  rgat_prep<<<1, F_OUT, 0, stream>>>(w, a, cs, BsrcP, BdstP);

  const int WPB = 8;  // waves per 256-thread block
  rgat_score<<<(node_tiles + WPB - 1) / WPB, 256, 0, stream>>>(h, BsrcP, Snode, N);
  rgat_score<<<(nrel_tiles + WPB - 1) / WPB, 256, 0, stream>>>(inputr, BdstP, Srel, NREL);

  // Grid-stride edge phase: 4096 blocks * 8 waves = 32768 waves (~18 edges ea)
  int eblocks = (E + WPB - 1) / WPB;
  if (eblocks > 4096) eblocks = 4096;
  rgat_edge<<<eblocks, 256, 0, stream>>>(A, h, inputr, cs, Snode, Srel,
                                         rowsum, (float*)d_out, E, N);

  rgat_norm<<<(out_size + 255) / 256, 256, 0, stream>>>((float*)d_out, rowsum, out_size);
}